// GCN_87479893885198
// MI455X (gfx1250) — compile-verified
//
#include <hip/hip_runtime.h>

#define FIN  32
#define EMB  64
#define FOUT 8
#define NB   64

typedef __attribute__((ext_vector_type(2))) float v2f;
typedef __attribute__((ext_vector_type(8))) float v8f;

// ---------------- degree / norm (computed once, reused by all 4 layers) -----
__global__ void k_deg_init(int* __restrict__ deg, int n) {
  int i = blockIdx.x * blockDim.x + threadIdx.x;
  if (i < n) deg[i] = 1;                       // self-loop contribution
}
__global__ void k_deg_edges(const int* __restrict__ dst, int* __restrict__ deg, int e) {
  int i = blockIdx.x * blockDim.x + threadIdx.x;
  if (i < e) atomicAdd(&deg[dst[i]], 1);
}
__global__ void k_dinv(const int* __restrict__ deg, float* __restrict__ dinv, int n) {
  int i = blockIdx.x * blockDim.x + threadIdx.x;
  if (i < n) dinv[i] = rsqrtf((float)deg[i]);  // deg >= 1 always
}
__global__ void k_norm(const int* __restrict__ src, const int* __restrict__ dst,
                       const float* __restrict__ dinv, float* __restrict__ norm, int e) {
  int i = blockIdx.x * blockDim.x + threadIdx.x;
  if (i < e) norm[i] = dinv[src[i]] * dinv[dst[i]];
}

// ---------------- fp32 WMMA GEMM: H = X*W ; ACC = dinv^2 * H (self loop) ----
// One wave computes a 16-row x 64-col strip. Full fp32 via V_WMMA_F32_16X16X4_F32.
// ISA 7.12.2 layouts (wave32):
//   A 16x4 : lane<16 -> M=lane, {v0,v1}=K{k0,k0+1}; lane>=16 -> M=lane-16, K{k0+2,k0+3}
//   B 4x16 : lane<16 -> N=lane, {v0,v1}=K{k0,k0+1}; lane>=16 -> N=lane-16, K{k0+2,k0+3}
//   C/D    : VGPR r -> row = r + 8*(lane>=16), col = lane&15
template <int K>
__global__ __launch_bounds__(256) void k_gemm_wmma(
    const float* __restrict__ X, const float* __restrict__ W,
    const float* __restrict__ dinv, float* __restrict__ H,
    float* __restrict__ ACC, int n)
{
  const int lane  = threadIdx.x & 31;
  const int strip = blockIdx.x * 8 + (threadIdx.x >> 5);  // 8 waves / block
  const int rowBase = strip * 16;
  if (rowBase >= n) return;                               // wave-uniform: EXEC stays all-1
  const int half = lane >> 4;
  const int l16  = lane & 15;

  v8f acc[4];
#pragma unroll
  for (int t = 0; t < 4; ++t)
#pragma unroll
    for (int r = 0; r < 8; ++r) acc[t][r] = 0.0f;

  const float* arow = X + (size_t)(rowBase + l16) * K + half * 2;
#pragma unroll
  for (int k0 = 0; k0 < K; k0 += 4) {
    v2f a;
    a.x = arow[k0];
    a.y = arow[k0 + 1];
    const int kb = k0 + half * 2;
    const float* b0 = W + (size_t)kb * EMB;   // row kb   of W [K x 64]
    const float* b1 = b0 + EMB;               // row kb+1
#pragma unroll
    for (int nt = 0; nt < 4; ++nt) {
      v2f b;
      b.x = b0[nt * 16 + l16];
      b.y = b1[nt * 16 + l16];
      acc[nt] = __builtin_amdgcn_wmma_f32_16x16x4_f32(
          /*neg_a=*/false, a, /*neg_b=*/false, b,
          /*c_mod=*/(short)0, acc[nt], /*reuse_a=*/false, /*reuse_b=*/false);
    }
  }

#pragma unroll
  for (int r = 0; r < 8; ++r) {
    const int row = rowBase + half * 8 + r;
    const float di = dinv[row];
    const float d2 = di * di;
#pragma unroll
    for (int nt = 0; nt < 4; ++nt) {
      const float v = acc[nt][r];
      const size_t idx = (size_t)row * EMB + nt * 16 + l16;
      H[idx]   = v;        // dense result (gathered by edges)
      ACC[idx] = d2 * v;   // accumulator pre-seeded with self-loop term
    }
  }
}

// ---------------- edge scatter: ACC[dst] += norm[e] * H[src] ----------------
// 16 lanes per edge, each lane owns a float4 feature chunk.
__global__ __launch_bounds__(256) void k_scatter(
    const int* __restrict__ src, const int* __restrict__ dst,
    const float* __restrict__ norm, const float* __restrict__ H,
    float* __restrict__ ACC, int e)
{
  int t = blockIdx.x * blockDim.x + threadIdx.x;
  int i = t >> 4;
  const int c = (t & 15) << 2;
  const int stride = (gridDim.x * blockDim.x) >> 4;
  for (; i < e; i += stride) {
    const int s = src[i], d = dst[i];
    const float w = norm[i];
    const float4 v = *(const float4*)(H + (size_t)s * EMB + c);
    float* o = ACC + (size_t)d * EMB + c;
    atomicAdd(o + 0, w * v.x);
    atomicAdd(o + 1, w * v.y);
    atomicAdd(o + 2, w * v.z);
    atomicAdd(o + 3, w * v.w);
  }
}

// ---------------- bias + ReLU (in place; result becomes next layer input) ---
__global__ void k_bias_relu(float* __restrict__ A, const float* __restrict__ bias, int total) {
  int t = blockIdx.x * blockDim.x + threadIdx.x;
  if (t < total) {
    float v = A[t] + bias[t & (EMB - 1)];
    A[t] = v > 0.0f ? v : 0.0f;
  }
}

// ---------------- segment pooling (LDS-staged sum/max/count) ----------------
__global__ void k_pool_init(float* __restrict__ gmax, float* __restrict__ gsum,
                            int* __restrict__ gcnt) {
  int i = blockIdx.x * blockDim.x + threadIdx.x;
  if (i < NB * EMB) { gmax[i] = 0.0f; gsum[i] = 0.0f; }
  if (i < NB) gcnt[i] = 0;
}
__global__ __launch_bounds__(256) void k_pool(
    const float* __restrict__ X, const int* __restrict__ batch,
    float* __restrict__ gmax, float* __restrict__ gsum, int* __restrict__ gcnt, int n)
{
  __shared__ float    ssum[NB * EMB];
  __shared__ unsigned smax[NB * EMB];   // post-ReLU values >= 0: uint order == float order
  __shared__ int      scnt[NB];
  for (int i = threadIdx.x; i < NB * EMB; i += blockDim.x) { ssum[i] = 0.0f; smax[i] = 0u; }
  if (threadIdx.x < NB) scnt[threadIdx.x] = 0;
  __syncthreads();

  const int f   = threadIdx.x & (EMB - 1);
  const int sub = threadIdx.x >> 6;     // 4 nodes in flight per block
  for (int node = blockIdx.x * 4 + sub; node < n; node += gridDim.x * 4) {
    const int b = batch[node];
    const float v = X[(size_t)node * EMB + f];
    atomicAdd(&ssum[b * EMB + f], v);                     // ds_add_f32
    atomicMax(&smax[b * EMB + f], __float_as_uint(v));    // ds_max_u32
    if (f == 0) atomicAdd(&scnt[b], 1);
  }
  __syncthreads();

  for (int i = threadIdx.x; i < NB * EMB; i += blockDim.x) {
    const float sv = ssum[i];
    if (sv != 0.0f) atomicAdd(&gsum[i], sv);
    const unsigned mv = smax[i];
    if (mv != 0u) atomicMax((unsigned*)gmax + i, mv);
  }
  if (threadIdx.x < NB) {
    const int cv = scnt[threadIdx.x];
    if (cv) atomicAdd(&gcnt[threadIdx.x], cv);
  }
}

// ---------------- final linear: out = [gmax | gsum/cnt] @ Wout + bout -------
__global__ void k_final(const float* __restrict__ gmax, const float* __restrict__ gsum,
                        const int* __restrict__ gcnt, const float* __restrict__ Wout,
                        const float* __restrict__ bout, float* __restrict__ out)
{
  const int t = threadIdx.x;
  if (t >= NB * FOUT) return;
  const int g = t >> 3, o = t & 7;
  float cnt = (float)gcnt[g];
  if (cnt < 1.0f) cnt = 1.0f;
  const float inv = 1.0f / cnt;
  float a = bout[o];
  const float* gm = gmax + g * EMB;
  const float* gs = gsum + g * EMB;
#pragma unroll 4
  for (int k = 0; k < EMB; ++k) a += gm[k] * Wout[k * FOUT + o];
#pragma unroll 4
  for (int k = 0; k < EMB; ++k) a += gs[k] * inv * Wout[(EMB + k) * FOUT + o];
  out[g * FOUT + o] = a;
}

// ---------------------------------------------------------------------------
extern "C" void kernel_launch(void* const* d_in, const int* in_sizes, int n_in,
                              void* d_out, int out_size, void* d_ws, size_t ws_size,
                              hipStream_t stream) {
  const float* x     = (const float*)d_in[0];
  const int*   ei    = (const int*)d_in[1];     // integer inputs arrive as int32
  const int*   batch = (const int*)d_in[2];
  const float* Ws[4] = {(const float*)d_in[3], (const float*)d_in[5],
                        (const float*)d_in[7], (const float*)d_in[9]};
  const float* bs[4] = {(const float*)d_in[4], (const float*)d_in[6],
                        (const float*)d_in[8], (const float*)d_in[10]};
  const float* Wout  = (const float*)d_in[11];
  const float* bout  = (const float*)d_in[12];
  float* out = (float*)d_out;

  const int n = in_sizes[0] / FIN;      // 100000 (multiple of 16)
  const int e = in_sizes[1] / 2;        // 1600000
  const int* src = ei;
  const int* dst = ei + e;

  // Carve scratch (256B-aligned): ~84 MB total.
  char* ws = (char*)d_ws;
  size_t off = 0;
  auto carve = [&](size_t bytes) -> char* {
    char* p = ws + off;
    off = (off + bytes + 255) & ~(size_t)255;
    return p;
  };
  int*   degI = (int*)  carve((size_t)n * 4);
  float* dinv = (float*)carve((size_t)n * 4);
  float* norm = (float*)carve((size_t)e * 4);
  float* bufH = (float*)carve((size_t)n * EMB * 4);
  float* p0   = (float*)carve((size_t)n * EMB * 4);
  float* p1   = (float*)carve((size_t)n * EMB * 4);
  float* gmax = (float*)carve((size_t)NB * EMB * 4);
  float* gsum = (float*)carve((size_t)NB * EMB * 4);
  int*   gcnt = (int*)  carve((size_t)NB * 4);
  (void)ws_size; (void)n_in; (void)out_size;

  const int B256 = 256;
  // degrees / norms (once)
  k_deg_init <<<(n + 255) / 256, B256, 0, stream>>>(degI, n);
  k_deg_edges<<<(e + 255) / 256, B256, 0, stream>>>(dst, degI, e);
  k_dinv     <<<(n + 255) / 256, B256, 0, stream>>>(degI, dinv, n);
  k_norm     <<<(e + 255) / 256, B256, 0, stream>>>(src, dst, dinv, norm, e);

  const int strips = (n + 15) / 16;
  const int gemmBlocks = (strips + 7) / 8;

  const float* cur = x;
  float* pp[2] = {p0, p1};
  for (int L = 0; L < 4; ++L) {
    float* accb = pp[L & 1];
    if (L == 0)
      k_gemm_wmma<FIN><<<gemmBlocks, B256, 0, stream>>>(cur, Ws[L], dinv, bufH, accb, n);
    else
      k_gemm_wmma<EMB><<<gemmBlocks, B256, 0, stream>>>(cur, Ws[L], dinv, bufH, accb, n);
    k_scatter<<<8192, B256, 0, stream>>>(src, dst, norm, bufH, accb, e);
    k_bias_relu<<<((size_t)n * EMB + 255) / 256, B256, 0, stream>>>(accb, bs[L], n * EMB);
    cur = accb;
  }

  k_pool_init<<<(NB * EMB + 255) / 256, B256, 0, stream>>>(gmax, gsum, gcnt);
  k_pool     <<<256, B256, 0, stream>>>(cur, batch, gmax, gsum, gcnt, n);
  k_final    <<<1, NB * FOUT, 0, stream>>>(gmax, gsum, gcnt, Wout, bout, out);
}